// TanhFixedPointLayer_6571299963072
// MI455X (gfx1250) — compile-verified
//
#include <hip/hip_runtime.h>
#include <cmath>

// CDNA5 / gfx1250: wave32, WMMA f32 16x16x4.
typedef __attribute__((ext_vector_type(2))) float v2f;
typedef __attribute__((ext_vector_type(8))) float v8f;

#define B_ROWS 262144
#define DDIM   64
#define ITERS  50
#define WROW   68   // padded LDS row stride (floats) -> conflict-free b64 reads
#define ZROW   68

__device__ __forceinline__ float fast_tanh(float v) {
#if defined(__gfx1250__) && __has_builtin(__builtin_amdgcn_tanhf)
  return __builtin_amdgcn_tanhf(v);   // v_tanh_f32 (gfx1250 TRANS op)
#else
  return tanhf(v);
#endif
}

__global__ __launch_bounds__(256) void tanh_fixed_point_kernel(
    const float* __restrict__ x, const float* __restrict__ W,
    float* __restrict__ out) {
  __shared__ float lds_w[DDIM * WROW];          // W, padded
  __shared__ float lds_z[8 * 16 * ZROW];        // per-wave z transpose scratch

  const int tid  = threadIdx.x;
  const int wave = tid >> 5;
  const int lane = tid & 31;
  const int half = lane >> 4;   // 0: lanes 0-15, 1: lanes 16-31
  const int lr   = lane & 15;

  // Stage W[d][k] into padded LDS (row-major, stride WROW).
  for (int i = tid; i < DDIM * DDIM; i += 256) {
    int d = i >> 6, k = i & 63;
    lds_w[d * WROW + k] = W[i];
  }
  __syncthreads();

  const int  chunk = blockIdx.x * 8 + wave;     // 16-row chunk per wave
  const long row0  = (long)chunk * 16;
  float* zbuf = &lds_z[wave * 16 * ZROW];       // wave-private

  // Load x tile in C/D fragment layout:
  //   xf[nt][j] = x[row0 + j + half*8][nt*16 + lr]
  v8f xf[4];
#pragma unroll
  for (int nt = 0; nt < 4; ++nt) {
#pragma unroll
    for (int j = 0; j < 8; ++j) {
      long r = row0 + j + half * 8;
      xf[nt][j] = x[r * DDIM + nt * 16 + lr];
    }
  }

  // Iteration 1 (z0 = 0):  z = tanh(x)
  v8f d[4];
#pragma unroll
  for (int nt = 0; nt < 4; ++nt)
#pragma unroll
    for (int j = 0; j < 8; ++j)
      d[nt][j] = fast_tanh(xf[nt][j]);

  // Remaining iterations: z <- tanh(z W^T + x), z kept on-chip.
  for (int it = 1; it < ITERS; ++it) {
    // C/D layout -> row-major [m][k] in wave-private LDS (in-order DS, no barrier).
#pragma unroll
    for (int nt = 0; nt < 4; ++nt)
#pragma unroll
      for (int j = 0; j < 8; ++j)
        zbuf[(j + half * 8) * ZROW + nt * 16 + lr] = d[nt][j];

    // A fragments: a[ks] = z[m=lr][k = ks*4 + half*2 .. +1]  (16x4 f32 A layout)
    v2f a[16];
#pragma unroll
    for (int ks = 0; ks < 16; ++ks)
      a[ks] = *(const v2f*)&zbuf[lr * ZROW + ks * 4 + half * 2];

#pragma unroll
    for (int nt = 0; nt < 4; ++nt) {
      v8f acc = xf[nt];   // C = x
#pragma unroll
      for (int ks = 0; ks < 16; ++ks) {
        // B fragment (4x16): b = W[n = nt*16+lr][k = ks*4 + half*2 .. +1]
        v2f b = *(const v2f*)&lds_w[(nt * 16 + lr) * WROW + ks * 4 + half * 2];
        acc = __builtin_amdgcn_wmma_f32_16x16x4_f32(
            /*neg_a=*/false, a[ks], /*neg_b=*/false, b,
            /*c_mod=*/(short)0, acc, /*reuse_a=*/false, /*reuse_b=*/false);
      }
#pragma unroll
      for (int j = 0; j < 8; ++j)
        d[nt][j] = fast_tanh(acc[j]);
    }
  }

  // Write final z (C/D layout -> global row-major).
#pragma unroll
  for (int nt = 0; nt < 4; ++nt)
#pragma unroll
    for (int j = 0; j < 8; ++j) {
      long r = row0 + j + half * 8;
      out[r * DDIM + nt * 16 + lr] = d[nt][j];
    }
}

extern "C" void kernel_launch(void* const* d_in, const int* in_sizes, int n_in,
                              void* d_out, int out_size, void* d_ws, size_t ws_size,
                              hipStream_t stream) {
  const float* x = (const float*)d_in[0];   // [B, D] f32
  const float* W = (const float*)d_in[1];   // [D, D] f32
  float* out = (float*)d_out;               // [B, D] f32

  dim3 grid(B_ROWS / (16 * 8));             // 2048 blocks, 8 waves each
  dim3 block(256);
  tanh_fixed_point_kernel<<<grid, block, 0, stream>>>(x, W, out);
}